// CPDecoding_5935644803430
// MI455X (gfx1250) — compile-verified
//
#include <hip/hip_runtime.h>
#include <cstdint>

// ---------------------------------------------------------------------------
// CP decoding (TensoRF line sampling), MI455X / gfx1250.
//
// line_coef [3][24][256] (72 KB) is staged into LDS once per workgroup via the
// Tensor Data Mover in iterate mode, which simultaneously TRANSPOSES it to a
// [r][d*24+c] layout with a 76-float row stride:
//   - per-iteration tile: x=1 element, y=24 components (stride 256 elems),
//     z=3 dims (stride 6144 elems)  -> 72 contiguous floats written to LDS
//   - 256 iterations, global_addr += 1 elem, lds_addr += 76 elems
// Row stride 76 (mod 64 = 12) => LDS bank conflicts only when sampled rows
// differ by a multiple of 16. Each (dim,row) slice is 24 floats, 16B-aligned,
// so the gather inner loop is pure ds_load_b128.
// ---------------------------------------------------------------------------

#define RES        256
#define NC         24
#define LDS_STRIDE 76                    // floats per r-row: 72 data + 4 pad
#define LDS_FLOATS (RES * LDS_STRIDE)    // 19456 floats = 77824 bytes
#define BLOCK      256
#define PPT        4                     // points per thread

typedef uint32_t v4u __attribute__((ext_vector_type(4)));
typedef int      v8i __attribute__((ext_vector_type(8)));
typedef int      v4i __attribute__((ext_vector_type(4)));

__device__ __forceinline__ float eval_point(const float* __restrict__ lds,
                                            float x, float y, float z) {
  // Reference stacks coords as (z, y, x): dim0 sampled at z, dim1 at y, dim2 at x.
  float p0 = (z + 1.0f) * 127.5f;
  float p1 = (y + 1.0f) * 127.5f;
  float p2 = (x + 1.0f) * 127.5f;
  float fl0 = floorf(p0), fl1 = floorf(p1), fl2 = floorf(p2);
  int i0 = min(max((int)fl0, 0), RES - 2);
  int i1 = min(max((int)fl1, 0), RES - 2);
  int i2 = min(max((int)fl2, 0), RES - 2);
  float w0 = p0 - fl0, w1 = p1 - fl1, w2 = p2 - fl2;

  const float4* a0 = (const float4*)(lds + i0 * LDS_STRIDE + 0 * NC);
  const float4* b0 = (const float4*)(lds + (i0 + 1) * LDS_STRIDE + 0 * NC);
  const float4* a1 = (const float4*)(lds + i1 * LDS_STRIDE + 1 * NC);
  const float4* b1 = (const float4*)(lds + (i1 + 1) * LDS_STRIDE + 1 * NC);
  const float4* a2 = (const float4*)(lds + i2 * LDS_STRIDE + 2 * NC);
  const float4* b2 = (const float4*)(lds + (i2 + 1) * LDS_STRIDE + 2 * NC);

  float sum = 0.0f;
#pragma unroll
  for (int q = 0; q < NC / 4; ++q) {
    float4 f0a = a0[q], f0b = b0[q];
    float4 f1a = a1[q], f1b = b1[q];
    float4 f2a = a2[q], f2b = b2[q];
    float g0x = fmaf(w0, f0b.x - f0a.x, f0a.x);
    float g0y = fmaf(w0, f0b.y - f0a.y, f0a.y);
    float g0z = fmaf(w0, f0b.z - f0a.z, f0a.z);
    float g0w = fmaf(w0, f0b.w - f0a.w, f0a.w);
    float g1x = fmaf(w1, f1b.x - f1a.x, f1a.x);
    float g1y = fmaf(w1, f1b.y - f1a.y, f1a.y);
    float g1z = fmaf(w1, f1b.z - f1a.z, f1a.z);
    float g1w = fmaf(w1, f1b.w - f1a.w, f1a.w);
    float g2x = fmaf(w2, f2b.x - f2a.x, f2a.x);
    float g2y = fmaf(w2, f2b.y - f2a.y, f2a.y);
    float g2z = fmaf(w2, f2b.z - f2a.z, f2a.z);
    float g2w = fmaf(w2, f2b.w - f2a.w, f2a.w);
    sum += g0x * g1x * g2x;
    sum += g0y * g1y * g2y;
    sum += g0z * g1z * g2z;
    sum += g0w * g1w * g2w;
  }
  return sum;
}

__global__ void __launch_bounds__(BLOCK)
cp_decode_kernel(const float* __restrict__ pts,
                 const float* __restrict__ line,
                 float* __restrict__ out, int npts) {
  extern __shared__ float lds[];

#if __has_builtin(__builtin_amdgcn_tensor_load_to_lds)
  if (threadIdx.x < 32u) {   // first wave issues one TDM op for the workgroup
    uint64_t gaddr = (uint64_t)(uintptr_t)line;
    uint32_t laddr = (uint32_t)(uintptr_t)lds;   // flat low 32 bits = LDS byte offset

    // D# group 0: count=1 | lds_addr | global_addr[56:0] | type=2
    v4u g0;
    g0[0] = 1u;
    g0[1] = laddr;
    g0[2] = (uint32_t)gaddr;
    g0[3] = ((uint32_t)(gaddr >> 32) & 0x01FFFFFFu) | (2u << 30);

    // D# group 1: data_size=4B(2), iterate_enable=1, wg_mask=0
    // tensor_dim0=256 (r), tensor_dim1=24 (c), tile=(1,24,3)
    // tensor_dim0_stride=256 (c step), tensor_dim1_stride=6144 (d step)
    v8i g1;
    g1[0] = (int)((2u << 16) | (1u << 19));   // data_size | iterate_enable
    g1[1] = (int)(256u << 16);                // tensor_dim0 lo16 in [63:48]
    g1[2] = (int)(24u << 16);                 // tensor_dim0 hi16=0 | tensor_dim1 lo16
    g1[3] = (int)(1u << 16);                  // tensor_dim1 hi16=0 | tile_dim0=1
    g1[4] = (int)(24u | (3u << 16));          // tile_dim1=24 | tile_dim2=3
    g1[5] = 256;                              // tensor_dim0_stride[31:0]
    g1[6] = (int)(6144u << 16);               // t0s[47:32]=0 | tensor_dim1_stride lo16
    g1[7] = 0;                                // tensor_dim1_stride[47:16]

    // D# group 2 (iterate mode): tensor_dim2=3, lds_inc=76 elems,
    // global_inc=1 elem, iterate_count=255 (=256 iterations over r)
    v4i g2;
    g2[0] = 3;
    g2[1] = LDS_STRIDE;
    g2[2] = 1;
    g2[3] = (int)(255u << 16);

    v4i g3 = {0, 0, 0, 0};
    v8i g4 = {0, 0, 0, 0, 0, 0, 0, 0};   // VADDR4 slot: unused, zeros

    __builtin_amdgcn_tensor_load_to_lds(g0, g1, g2, g3, g4, 0);
    __builtin_amdgcn_s_wait_tensorcnt(0);
  }
#else
  // Fallback: cooperative transposed fill (coalesced global reads).
  for (int e = (int)threadIdx.x; e < 3 * NC * RES; e += BLOCK) {
    int r  = e & (RES - 1);
    int dc = e >> 8;                  // d*24 + c
    lds[r * LDS_STRIDE + dc] = line[e];
  }
#endif
  __syncthreads();

  int t = (int)blockIdx.x * BLOCK + (int)threadIdx.x;
  long base = (long)t * PPT;
  if (base >= npts) return;

  if (base + PPT <= (long)npts) {
    const float4* p4 = (const float4*)pts;   // base*3 floats = 48B*t, 16B aligned
    float4 A = p4[3 * t + 0];
    float4 B = p4[3 * t + 1];
    float4 C = p4[3 * t + 2];
    float4 r;
    r.x = eval_point(lds, A.x, A.y, A.z);
    r.y = eval_point(lds, A.w, B.x, B.y);
    r.z = eval_point(lds, B.z, B.w, C.x);
    r.w = eval_point(lds, C.y, C.z, C.w);
    ((float4*)out)[t] = r;
  } else {
    for (int j = 0; j < PPT; ++j) {
      long p = base + j;
      if (p < (long)npts) {
        out[p] = eval_point(lds, pts[3 * p + 0], pts[3 * p + 1], pts[3 * p + 2]);
      }
    }
  }
}

extern "C" void kernel_launch(void* const* d_in, const int* in_sizes, int n_in,
                              void* d_out, int out_size, void* d_ws, size_t ws_size,
                              hipStream_t stream) {
  const float* pts  = (const float*)d_in[0];   // [N,3] float32
  const float* line = (const float*)d_in[1];   // [3,24,256] float32
  float* out = (float*)d_out;                  // [N] float32

  int npts = in_sizes[0] / 3;
  int nthreads = (npts + PPT - 1) / PPT;
  int blocks = (nthreads + BLOCK - 1) / BLOCK;
  size_t shmem = (size_t)LDS_FLOATS * sizeof(float);   // 77824 B < 320 KB/WGP

  cp_decode_kernel<<<dim3(blocks), dim3(BLOCK), shmem, stream>>>(pts, line, out, npts);
}